// DeltaRecurrentUpdate_83004537962791
// MI455X (gfx1250) — compile-verified
//
#include <hip/hip_runtime.h>
#include <hip/hip_bf16.h>

// ---------------------------------------------------------------------------
// DeltaNet-style recurrence on MI455X (gfx1250), fp32 WMMA throughout.
//   B=16, T=2048, H=1024, R=64
// Kernel 1: proj_gemm  -> k = silu(hs @ Wk^T) [B*T,1024], v = hs @ Wv^T [B*T,64]
// Kernel 2: beta_kernel -> beta = sigmoid(hs @ Wg^T) [B*T]
// Kernel 3: delta_chunk -> exact chunked (WY-form) delta rule, S in LDS (256KB),
//           K-tile staging via GLOBAL_LOAD_ASYNC_TO_LDS_B128 (ASYNCcnt DMA)
// ---------------------------------------------------------------------------

typedef __attribute__((ext_vector_type(2))) float v2f;
typedef __attribute__((ext_vector_type(8))) float v8f;

#define B_  16
#define T_  2048
#define H_  1024
#define R_  64
#define CHK 16      // timestep chunk
#define HT  256     // H tile in recurrence
#define KSP 260     // Ks padded row stride (floats): bank-conflict-free frags, 16B-aligned rows
#define PBP 68      // P/U padded row stride
#define ABP 20      // A (K K^T) padded row stride

__device__ __forceinline__ v8f wmma4(v2f a, v2f b, v8f c) {
  // V_WMMA_F32_16X16X4_F32 : D[16x16] = A[16x4] * B[4x16] + C
  return __builtin_amdgcn_wmma_f32_16x16x4_f32(
      /*neg_a=*/false, a, /*neg_b=*/false, b,
      /*c_mod=*/(short)0, c, /*reuse_a=*/false, /*reuse_b=*/false);
}

__device__ __forceinline__ float sigmoidf_(float x) {
  // v_rcp_f32 instead of IEEE division chain (~1 ulp; fine for sigmoid)
  return __builtin_amdgcn_rcpf(1.f + __expf(-x));
}

// Async DMA: 16 bytes per lane, global -> LDS, tracked by ASYNCcnt.
// Per CDNA5 ISA (GV mode): copies 16B from the global address to the LDS address held in vdst
__device__ __forceinline__ void async_lds_load_b128(unsigned lds_off,
                                                    const void* gaddr) {
  unsigned long long ga = (unsigned long long)(uintptr_t)gaddr;
  asm volatile("global_load_async_to_lds_b128 %0, %1, off"
               :: "v"(lds_off), "v"(ga) : "memory");
}
__device__ __forceinline__ void wait_asynccnt0() {
  asm volatile("s_wait_asynccnt 0x0" ::: "memory");
}

// ---------------------------------------------------------------------------
// Kernel 1: C[M,N] = act(A[M,K] @ W[N,K]^T). Row-major, K contiguous for both.
// Block = 128 threads (4 waves), 64x64 output tile, BLOCK_K = 16.
// Wave w owns N columns [16w,16w+16); computes 4 M-subtiles (16x16 each).
// ---------------------------------------------------------------------------
__global__ void __launch_bounds__(128)
proj_gemm_kernel(const float* __restrict__ A, const float* __restrict__ W,
                 float* __restrict__ C, int K, int ldc, int act) {
  __shared__ float As[64][20];   // [m][k], row stride 20 (80B: aligned + conflict-free)
  __shared__ float Bs[64][20];   // [n][k]
  const int tid  = threadIdx.x;
  const int lane = tid & 31;
  const int w    = tid >> 5;
  const int l15  = lane & 15;
  const int kh   = (lane >> 4) << 1;   // 0 for lanes 0-15, 2 for lanes 16-31
  const int m0   = blockIdx.x * 64;
  const int n0   = blockIdx.y * 64;

  v8f c[4] = {};

  for (int kt = 0; kt < K; kt += 16) {
    __syncthreads();
#pragma unroll
    for (int i = 0; i < 2; ++i) {      // 64 rows x 4 float4 = 256 float4 / 128 thr
      int f = i * 128 + tid;
      int row = f >> 2, kc = f & 3;
      *(float4*)&As[row][kc * 4] =
          *(const float4*)(A + (size_t)(m0 + row) * K + kt + kc * 4);
      *(float4*)&Bs[row][kc * 4] =
          *(const float4*)(W + (size_t)(n0 + row) * K + kt + kc * 4);
    }
    __syncthreads();
#pragma unroll
    for (int k0 = 0; k0 < 16; k0 += 4) {
      // B frag: lane L<16 -> rows k0,k0+1 ; L>=16 -> rows k0+2,k0+3 (col = n)
      v2f bf;
      bf.x = Bs[16 * w + l15][k0 + kh];
      bf.y = Bs[16 * w + l15][k0 + kh + 1];
#pragma unroll
      for (int m = 0; m < 4; ++m) {
        // A frag: lane L<16 -> row M=L, K=k0,k0+1 ; L>=16 -> row M=L-16, K=k0+2,k0+3
        v2f af;
        af.x = As[16 * m + l15][k0 + kh];
        af.y = As[16 * m + l15][k0 + kh + 1];
        c[m] = wmma4(af, bf, c[m]);
      }
    }
  }
  // C/D layout: VGPR r -> rows r (lanes 0-15) and r+8 (lanes 16-31), col = lane%16
#pragma unroll
  for (int m = 0; m < 4; ++m) {
#pragma unroll
    for (int r = 0; r < 8; ++r) {
      int row = m0 + 16 * m + r + 8 * (lane >> 4);
      int col = n0 + 16 * w + l15;
      float val = c[m][r];
      if (act) val = val * sigmoidf_(val);   // silu
      C[(size_t)row * ldc + col] = val;
    }
  }
}

// ---------------------------------------------------------------------------
// Kernel 2: beta[row] = sigmoid(dot(hs[row,:], Wg)). One wave32 per row.
// ---------------------------------------------------------------------------
__global__ void __launch_bounds__(256)
beta_kernel(const float* __restrict__ hs, const float* __restrict__ wg,
            float* __restrict__ out) {
  int row  = blockIdx.x * 8 + (threadIdx.x >> 5);
  int lane = threadIdx.x & 31;
  const float4* hp = (const float4*)(hs + (size_t)row * H_);
  const float4* wp = (const float4*)wg;
  float acc = 0.f;
#pragma unroll
  for (int i = 0; i < 8; ++i) {     // 1024 / (32 lanes * 4) = 8
    float4 h4 = hp[lane + 32 * i];
    float4 w4 = wp[lane + 32 * i];
    acc += h4.x * w4.x + h4.y * w4.y + h4.z * w4.z + h4.w * w4.w;
  }
#pragma unroll
  for (int off = 16; off > 0; off >>= 1) acc += __shfl_xor(acc, off, 32);
  if (lane == 0) out[row] = sigmoidf_(acc);
}

// ---------------------------------------------------------------------------
// Kernel 3: chunked delta rule. One block per batch. S[1024][64] f32 in LDS.
// Per chunk (C=16):  P = K@S, A = K K^T  (WMMA over H, fp32)
//                    U : forward substitution (I + diag(b) tril(A)) U = diag(b)(V-P)
//                    S += K^T U          (WMMA outer-product update)
// ---------------------------------------------------------------------------
__global__ void __launch_bounds__(256)
delta_chunk_kernel(const float* __restrict__ kall, const float* __restrict__ vall,
                   const float* __restrict__ ball, const float* __restrict__ prev,
                   float* __restrict__ out) {
  extern __shared__ float smem[];
  float* S  = smem;                     // H_*R_            = 65536 f
  float* Ks = S  + H_ * R_;             // CHK*KSP          =  4160 f
  float* Pb = Ks + CHK * KSP;           // CHK*PBP          =  1088 f
  float* Ub = Pb + CHK * PBP;           // CHK*PBP          =  1088 f
  float* Vb = Ub + CHK * PBP;           // CHK*R_           =  1024 f
  float* Ab = Vb + CHK * R_;            // CHK*ABP          =   320 f
  float* Bt = Ab + CHK * ABP;           //                  =    16 f

  const int b    = blockIdx.x;
  const int tid  = threadIdx.x;
  const int lane = tid & 31;
  const int wv   = tid >> 5;
  const int l15  = lane & 15;
  const int kh   = (lane >> 4) << 1;    // 0 / 2
  const int hh   = (lane >> 4) << 3;    // 0 / 8 (C-frag row offset)

  // Load S <- prev_cache[b]
  {
    const float4* pc = (const float4*)(prev + (size_t)b * H_ * R_);
    float4* S4 = (float4*)S;
    for (int i = tid; i < H_ * R_ / 4; i += 256) S4[i] = pc[i];
  }

  const float* kb = kall + (size_t)b * T_ * H_;
  const float* vb = vall + (size_t)b * T_ * R_;
  const float* bb = ball + (size_t)b * T_;

  for (int ch = 0; ch < T_ / CHK; ++ch) {
    const int t0 = ch * CHK;
    __syncthreads();
    // Load V chunk and beta chunk
    {
      const float4* vsrc = (const float4*)(vb + (size_t)t0 * R_);
      ((float4*)Vb)[tid] = vsrc[tid];        // 16*64 = 1024 floats = 256 float4
      if (tid < CHK) Bt[tid] = bb[t0 + tid];
    }

    // ---- Phase 1: P = K@S (waves 0-3), A = K K^T (wave 4), over H tiles ----
    v8f accP = {};
    v8f accA = {};
    for (int ht = 0; ht < H_ / HT; ++ht) {
      __syncthreads();                       // prior Ks consumers retired
      // Stage K chunk tile Ks[16][256] via async DMA: 1024 x b128 / 256 thr
#pragma unroll
      for (int i = 0; i < 4; ++i) {
        int f = i * 256 + tid;
        int row = f >> 6, c4 = f & 63;
        async_lds_load_b128(
            (unsigned)(uintptr_t)&Ks[row * KSP + c4 * 4],
            kb + (size_t)(t0 + row) * H_ + ht * HT + c4 * 4);
      }
      wait_asynccnt0();
      __syncthreads();
      if (wv < 4) {
        const int col = 16 * wv + l15;       // column of R
        for (int k0 = 0; k0 < HT; k0 += 4) {
          v2f af;                            // A = K_chunk [16 x H]
          af.x = Ks[l15 * KSP + k0 + kh];
          af.y = Ks[l15 * KSP + k0 + kh + 1];
          const int h = ht * HT + k0 + kh;   // B = S [H x R]
          v2f bf;
          bf.x = S[h * R_ + col];
          bf.y = S[(h + 1) * R_ + col];
          accP = wmma4(af, bf, accP);
        }
      } else if (wv == 4) {
        for (int k0 = 0; k0 < HT; k0 += 4) {
          v2f af;
          af.x = Ks[l15 * KSP + k0 + kh];
          af.y = Ks[l15 * KSP + k0 + kh + 1];
          // B fragment of K^T is bit-identical to A fragment of K for 16x16x4
          accA = wmma4(af, af, accA);
        }
      }
    }
    __syncthreads();
    if (wv < 4) {
#pragma unroll
      for (int r = 0; r < 8; ++r)
        Pb[(r + hh) * PBP + 16 * wv + l15] = accP[r];
    } else if (wv == 4) {
#pragma unroll
      for (int r = 0; r < 8; ++r)
        Ab[(r + hh) * ABP + l15] = accA[r];
    }
    __syncthreads();

    // ---- Phase 2: forward substitution. Thread j owns column j of U. ----
    if (tid < 64) {
      float u[CHK];
#pragma unroll
      for (int t = 0; t < CHK; ++t) {
        float acc = Vb[t * R_ + tid] - Pb[t * PBP + tid];
#pragma unroll
        for (int s = 0; s < CHK; ++s)
          if (s < t) acc -= Ab[t * ABP + s] * u[s];
        u[t] = Bt[t] * acc;
      }
#pragma unroll
      for (int t = 0; t < CHK; ++t) Ub[t * PBP + tid] = u[t];
    }
    __syncthreads();

    // ---- Phase 3: S += K^T @ U, tiled over H. All 8 waves. ----
    for (int ht = 0; ht < H_ / HT; ++ht) {
#pragma unroll
      for (int i = 0; i < 4; ++i) {          // restage K tile (async DMA)
        int f = i * 256 + tid;
        int row = f >> 6, c4 = f & 63;
        async_lds_load_b128(
            (unsigned)(uintptr_t)&Ks[row * KSP + c4 * 4],
            kb + (size_t)(t0 + row) * H_ + ht * HT + c4 * 4);
      }
      wait_asynccnt0();
      __syncthreads();
#pragma unroll
      for (int i = 0; i < 8; ++i) {          // 16 M-subtiles x 4 N-subtiles / 8 waves
        int idx = wv * 8 + i;
        int ms = idx >> 2, ns = idx & 3;
        int hloc = 16 * ms + l15;
        int h0   = ht * HT + 16 * ms;
        int colr = 16 * ns + l15;
        v8f cS;
#pragma unroll
        for (int r = 0; r < 8; ++r) cS[r] = S[(h0 + r + hh) * R_ + colr];
#pragma unroll
        for (int k0 = 0; k0 < CHK; k0 += 4) {
          v2f af;                            // A = K^T [H x 16]
          af.x = Ks[(k0 + kh) * KSP + hloc];
          af.y = Ks[(k0 + kh + 1) * KSP + hloc];
          v2f bf;                            // B = U [16 x 64]
          bf.x = Ub[(k0 + kh) * PBP + colr];
          bf.y = Ub[(k0 + kh + 1) * PBP + colr];
          cS = wmma4(af, bf, cS);
        }
#pragma unroll
        for (int r = 0; r < 8; ++r) S[(h0 + r + hh) * R_ + colr] = cS[r];
      }
      __syncthreads();
    }
  }

  // Store S -> out[b]
  __syncthreads();
  {
    float4* op = (float4*)(out + (size_t)b * H_ * R_);
    const float4* S4 = (const float4*)S;
    for (int i = tid; i < H_ * R_ / 4; i += 256) op[i] = S4[i];
  }
}

// ---------------------------------------------------------------------------
extern "C" void kernel_launch(void* const* d_in, const int* in_sizes, int n_in,
                              void* d_out, int out_size, void* d_ws, size_t ws_size,
                              hipStream_t stream) {
  (void)in_sizes; (void)n_in; (void)out_size; (void)ws_size;
  const float* hs   = (const float*)d_in[0];  // [B,T,H]
  const float* prev = (const float*)d_in[1];  // [B,H,R]
  const float* Wk   = (const float*)d_in[2];  // [H,H]
  const float* Wv   = (const float*)d_in[3];  // [R,H]
  const float* Wg   = (const float*)d_in[4];  // [1,H]
  float* out = (float*)d_out;                 // [B,H,R]

  float* kall = (float*)d_ws;                       // [B*T, H]  f32
  float* vall = kall + (size_t)B_ * T_ * H_;        // [B*T, R]  f32
  float* ball = vall + (size_t)B_ * T_ * R_;        // [B*T]     f32

  const int M = B_ * T_;                            // 32768

  // k = silu(hs @ Wk^T)
  proj_gemm_kernel<<<dim3(M / 64, H_ / 64), dim3(128), 0, stream>>>(
      hs, Wk, kall, H_, H_, /*act=*/1);
  // v = hs @ Wv^T
  proj_gemm_kernel<<<dim3(M / 64, R_ / 64), dim3(128), 0, stream>>>(
      hs, Wv, vall, H_, R_, /*act=*/0);
  // beta = sigmoid(hs @ Wg^T)
  beta_kernel<<<dim3(M / 8), dim3(256), 0, stream>>>(hs, Wg, ball);

  // chunked delta-rule recurrence; S resident in LDS
  size_t lds_bytes = (size_t)(H_ * R_ + CHK * KSP + 2 * CHK * PBP +
                              CHK * R_ + CHK * ABP + CHK) * sizeof(float);
  delta_chunk_kernel<<<dim3(B_), dim3(256), lds_bytes, stream>>>(
      kall, vall, ball, prev, out);
}